// quantizer_channel_83648783057544
// MI455X (gfx1250) — compile-verified
//
#include <hip/hip_runtime.h>
#include <hip/hip_bf16.h>
#include <stdint.h>

// ---------------------------------------------------------------------------
// Soft quantizer + bit-flip channel (elementwise, memory/VALU bound).
// CDNA5 path: TDM tensor_load_to_lds stages x tiles into LDS, s_wait_tensorcnt.
// Threefry lanes paired: thread handles elements (i, i+M) sharing PRNG blocks.
// Vectorized: each thread owns 4 contiguous elements/side -> b128 LDS + stores.
// ---------------------------------------------------------------------------

#define BLOCK 256
#define VEC   4                   // contiguous elements per thread per side
#define CHUNK (BLOCK * VEC)       // 1024 elements per side per block
#define KC    16
#define LOG2E 1.4426950408889634f

typedef uint32_t u32;

typedef unsigned int __attribute__((ext_vector_type(4))) u32x4;
typedef int          __attribute__((ext_vector_type(4))) i32x4;
typedef int          __attribute__((ext_vector_type(8))) i32x8;
typedef float        __attribute__((ext_vector_type(4))) f32x4;

#if defined(__HIP_DEVICE_COMPILE__) && __has_builtin(__builtin_amdgcn_tensor_load_to_lds) && __has_builtin(__builtin_amdgcn_s_wait_tensorcnt)
#define HAVE_TDM 1
#else
#define HAVE_TDM 0
#endif

#if HAVE_TDM
// 1D tile DMA: global -> LDS, nelem fp32 elements (nelem <= 65535).
// D# layout per CDNA5 ISA ch.8 (group0 128b, group1 256b; groups 2/3 zero).
// clang-23 toolchain: 6-arg builtin (g0, g1, g2, g3, g4, cpol).
__device__ __forceinline__ void tdm_load_row(u32 lds_off, const void* gptr, u32 nelem) {
  unsigned long long ga = (unsigned long long)(uintptr_t)gptr;
  u32x4 g0;
  g0.x = 1u;                                             // count=1 (valid), user desc
  g0.y = lds_off;                                        // lds_addr (bytes)
  g0.z = (u32)ga;                                        // global_addr[31:0]
  g0.w = (u32)((ga >> 32) & 0x01FFFFFFu) | (2u << 30);   // global_addr[56:32] | type=2
  i32x8 g1;
  g1[0] = (int)(2u << 16);                               // data_size=4B; mask/barrier/iter/pad=0
  g1[1] = (int)((nelem & 0xFFFFu) << 16);                // tensor_dim0[15:0]
  g1[2] = (int)(((nelem >> 16) & 0xFFFFu) | (1u << 16)); // tensor_dim0[31:16], tensor_dim1=1
  g1[3] = (int)((nelem & 0xFFFFu) << 16);                // tensor_dim1 hi=0, tile_dim0=nelem
  g1[4] = 0;                                             // tile_dim1=0, tile_dim2=0
  g1[5] = (int)nelem;                                    // tensor_dim0_stride[31:0]
  g1[6] = 0;
  g1[7] = 0;
  i32x4 z4 = {0, 0, 0, 0};                               // groups 2/3 unused (<=2D tensor)
  i32x8 z8 = {0, 0, 0, 0, 0, 0, 0, 0};                   // extra operand, zero-filled
  __builtin_amdgcn_tensor_load_to_lds(g0, g1, z4, z4, z8, 0);
}
#endif

// ---- threefry2x32 with key = jax.random.key(42) -> (0, 42) -----------------
#define TF_R4(a,b,c,d)                                                  \
  x0 += x1; x1 = __builtin_rotateleft32(x1, a); x1 ^= x0;               \
  x0 += x1; x1 = __builtin_rotateleft32(x1, b); x1 ^= x0;               \
  x0 += x1; x1 = __builtin_rotateleft32(x1, c); x1 ^= x0;               \
  x0 += x1; x1 = __builtin_rotateleft32(x1, d); x1 ^= x0;

__device__ __forceinline__ void threefry_0_42(u32 x0, u32 x1, u32& y0, u32& y1) {
  const u32 ks0 = 0u, ks1 = 42u, ks2 = 0x1BD11BDAu ^ 42u;
  x0 += ks0; x1 += ks1;
  TF_R4(13, 15, 26, 6)   x0 += ks1; x1 += ks2 + 1u;
  TF_R4(17, 29, 16, 24)  x0 += ks2; x1 += ks0 + 2u;
  TF_R4(13, 15, 26, 6)   x0 += ks0; x1 += ks1 + 3u;
  TF_R4(17, 29, 16, 24)  x0 += ks1; x1 += ks2 + 4u;
  TF_R4(13, 15, 26, 6)   x0 += ks2; x1 += ks0 + 5u;
  y0 = x0; y1 = x1;
}

// bernoulli(0.01): jax uniform u = fl((bits>>9) * 2^-23); u < 0.01f
// <=> (bits>>9) < 0.01f*2^23 = 83886.078125  <=> (bits>>9) < 83887 (exact).
__device__ __forceinline__ u32 bern001(u32 bits) {
  return ((bits >> 9) < 83887u) ? 1u : 0u;
}

// softmax-weighted center + argmin index.
// t_k = log2e*(20*c_k*x - 10*c_k^2); argmax t_k == argmin |x-c_k|;
// smx = 2^(t_k - max) / sum (softmax is shift-invariant, -10x^2 term dropped).
__device__ __forceinline__ void soft_quant(float x,
                                           const float* __restrict__ ca,
                                           const float* __restrict__ cb,
                                           const float* __restrict__ cc,
                                           float& wsoft, u32& idx) {
  float t[KC];
#pragma unroll
  for (int k = 0; k < KC; ++k) t[k] = __builtin_fmaf(ca[k], x, cb[k]);
  float mx = t[0]; idx = 0u;
#pragma unroll
  for (int k = 1; k < KC; ++k)
    if (t[k] > mx) { mx = t[k]; idx = (u32)k; }
  float den = 0.0f, num = 0.0f;
#pragma unroll
  for (int k = 0; k < KC; ++k) {
    float e = __builtin_amdgcn_exp2f(t[k] - mx);
    den += e;
    num = __builtin_fmaf(cc[k], e, num);
  }
  wsoft = num * __builtin_amdgcn_rcpf(den);
}

__global__ __launch_bounds__(BLOCK) void qchan_soft_kernel(
    const float* __restrict__ x, const float* __restrict__ center,
    float* __restrict__ out, int M /* = N/2 */) {
  __shared__ __align__(16) float sLo[CHUNK];
  __shared__ __align__(16) float sHi[CHUNK];
  __shared__ float sC[KC];

  const int base = blockIdx.x * CHUNK;
  const int tid  = threadIdx.x;

  if (tid < KC) sC[tid] = center[tid];

#if HAVE_TDM
  if (tid < 32) {  // wave 0 only: TDM issue is per-wave (EXEC ignored)
    tdm_load_row((u32)(uintptr_t)(void*)sLo, x + base,     (u32)CHUNK);
    tdm_load_row((u32)(uintptr_t)(void*)sHi, x + base + M, (u32)CHUNK);
    __builtin_amdgcn_s_wait_tensorcnt(0);
  }
#else
#pragma unroll
  for (int it = 0; it < VEC; ++it) {
    int l = it * BLOCK + tid;
    int i = base + l;
    sLo[l] = (i < M) ? x[i]     : 0.0f;
    sHi[l] = (i < M) ? x[i + M] : 0.0f;
  }
#endif
  __syncthreads();

  // per-thread center coefficients (constant indices after unroll -> registers)
  float ca[KC], cb[KC], cc[KC];
#pragma unroll
  for (int k = 0; k < KC; ++k) {
    float c = sC[k];
    cc[k] = c;
    ca[k] = (20.0f * LOG2E) * c;
    cb[k] = (-10.0f * LOG2E) * c * c;
  }

  const u32 half = 4u * (u32)M;  // n/2 where n = 4*N = 8*M noise bits
  const int t4 = tid * VEC;
  const int i0 = base + t4;      // first low element of this thread

  // b128 LDS reads: 4 contiguous fp32 per side per thread
  f32x4 vLo = *(const f32x4*)(sLo + t4);
  f32x4 vHi = *(const f32x4*)(sHi + t4);
  f32x4 oLo, oHi;

#pragma unroll
  for (int e = 0; e < VEC; ++e) {
    const int i = i0 + e;

    float wLo, wHi; u32 iLo, iHi;
    soft_quant(vLo[e], ca, cb, cc, wLo, iLo);
    soft_quant(vHi[e], ca, cb, cc, wHi, iHi);

    // 4 threefry blocks serve BOTH paired elements (lane0 -> i, lane1 -> i+M)
    u32 nLo = 0u, nHi = 0u;
#pragma unroll
    for (u32 j = 0u; j < 4u; ++j) {
      u32 b = 4u * (u32)i + j;
      u32 y0, y1;
      threefry_0_42(b, b + half, y0, y1);
      nLo |= bern001(y0) << (3u - j);
      nHi |= bern001(y1) << (3u - j);
    }

    float qLo = sC[iLo ^ nLo];   // W_noisy via LDS gather
    float qHi = sC[iHi ^ nHi];

    oLo[e] = wLo + (qLo - wLo);  // == W_noisy, expression kept per reference
    oHi[e] = wHi + (qHi - wHi);
  }

  // b128 coalesced stores (512B per wave per instruction)
  if (i0 + (VEC - 1) < M) {
    *(f32x4*)(out + i0)     = oLo;
    *(f32x4*)(out + i0 + M) = oHi;
  } else {
#pragma unroll
    for (int e = 0; e < VEC; ++e) {
      if (i0 + e < M) {
        out[i0 + e]     = oLo[e];
        out[i0 + e + M] = oHi[e];
      }
    }
  }
}

extern "C" void kernel_launch(void* const* d_in, const int* in_sizes, int n_in,
                              void* d_out, int out_size, void* d_ws, size_t ws_size,
                              hipStream_t stream) {
  const float* x      = (const float*)d_in[0];
  const float* center = (const float*)d_in[1];
  float* out          = (float*)d_out;

  int N = in_sizes[0];         // 32*256*32*32 = 8,388,608
  int M = N / 2;               // pairing offset (4,194,304)
  int blocks = (M + CHUNK - 1) / CHUNK;  // 4096

  qchan_soft_kernel<<<dim3(blocks), dim3(BLOCK), 0, stream>>>(x, center, out, M);
}